// TermME_13546326851884
// MI455X (gfx1250) — compile-verified
//
#include <hip/hip_runtime.h>
#include <hip/hip_bf16.h>
#include <math.h>

// ---- model constants (match reference) ----
#define PROJ_LEN      277
#define EXPENSE_ACQ   300.0f
#define EXPENSE_MAINT 60.0f
#define MORT_ROWS     100
#define MORT_COLS     6

// ---- workspace layout (floats) ----
#define MORT_N    (MORT_ROWS * MORT_COLS)   // 600 : monthly mortality per table cell
#define MORT_OFF  0
#define DE_OFF    (MORT_OFF + MORT_N)       // 600 : float2[277] = (disc_t, disc_t*expM_t)
#define LAPSE_OFF (DE_OFF + 2 * PROJ_LEN)   // 1154: monthly lapse factor per duration-year
#define LAPSE_N   48
#define TAB_N     (LAPSE_OFF + LAPSE_N)     // 1202
#define PART_OFF  1216                      // per-block partial sums start here

typedef __attribute__((ext_vector_type(2))) float v2f;
typedef __attribute__((ext_vector_type(8))) float v8f;

// --------------------------------------------------------------------------
// Kernel 1: precompute all transcendental-dependent tables (one tiny block).
//   mortM[i]  = 1-(1-q_i)^(1/12)
//   de[t]     = (disc_t, disc_t * (60/12)*1.01^(t/12))
//   lapseM[d] = 1-(1-max(0.1-0.02d,0.02))^(1/12)
// --------------------------------------------------------------------------
__global__ void tables_kernel(const float* __restrict__ mort_table,    // [100*6]
                              const float* __restrict__ disc_rate_ann, // [30]
                              float* __restrict__ ws) {
  const float twelfth = 1.0f / 12.0f;
  for (int i = threadIdx.x; i < MORT_N; i += blockDim.x) {
    float q = mort_table[i];
    ws[MORT_OFF + i] = 1.0f - powf(1.0f - q, twelfth);
  }
  for (int t = threadIdx.x; t < PROJ_LEN; t += blockDim.x) {
    float tf = (float)t;
    float r = disc_rate_ann[t / 12 + 1];
    float disc = powf(1.0f + r, -tf * twelfth);
    float expM = (EXPENSE_MAINT * twelfth) * powf(1.01f, tf * twelfth);
    ws[DE_OFF + 2 * t]     = disc;
    ws[DE_OFF + 2 * t + 1] = disc * expM;
  }
  for (int d = threadIdx.x; d < LAPSE_N; d += blockDim.x) {
    float lr = fmaxf(0.1f - 0.02f * (float)d, 0.02f);
    ws[LAPSE_OFF + d] = 1.0f - powf(1.0f - lr, twelfth);
  }
}

// --------------------------------------------------------------------------
// Kernel 2: one thread per policy. Year-chunked recurrence:
//   bef_{t+1} = bef_t*(1-qm)*(1-lf)          (geometric within a year)
//   acc      += bef_t*(coef*disc_t - discE_t), coef = P - SA*qm (per year)
// t=0 handled outside the loop (new-business/commission terms).
// Deterministic per-block tree reduction -> partials.
// --------------------------------------------------------------------------
__global__ void __launch_bounds__(256)
project_kernel(const float* __restrict__ premium_pp,
               const float* __restrict__ sum_assured,
               const float* __restrict__ policy_count,
               const int*   __restrict__ duration_mth,
               const int*   __restrict__ age_at_entry,
               const int*   __restrict__ policy_term,
               const float* __restrict__ ws,
               float* __restrict__ partials,
               int n) {
  __shared__ float s_tab[TAB_N];
  __shared__ float s_red[256];

  for (int j = threadIdx.x; j < TAB_N; j += blockDim.x) s_tab[j] = ws[j];
  __syncthreads();

  const float*  s_mort  = s_tab + MORT_OFF;
  const float2* s_de    = (const float2*)(s_tab + DE_OFF);   // byte off 2400, 8B aligned
  const float*  s_lapse = s_tab + LAPSE_OFF;

  int i = blockIdx.x * blockDim.x + threadIdx.x;
  float acc = 0.0f;

  if (i < n) {
    const float P      = premium_pp[i];
    const float SA     = sum_assured[i];
    const float pc     = policy_count[i];
    const int   dm0    = duration_mth[i];
    const int   age0   = age_at_entry[i];
    const int   term12 = policy_term[i] * 12;

    // months with non-zero cash flow: t in [0, t_end)
    int rem   = term12 - dm0;                       // >= 1 by construction
    int t_end = rem < PROJ_LEN ? rem : PROJ_LEN;

    // ---- t = 0 (only month where new business / commission can occur) ----
    // bef_decr(0) = pc whether dm0==0 (all new biz) or dm0>0 (all in force).
    int dur0 = dm0 / 12;
    int dc0  = dur0 < 5 ? dur0 : 5;
    float qm = s_mort[(age0 + dur0 - 18 - dc0) * MORT_COLS + dc0];
    float lf = s_lapse[dur0 < (LAPSE_N - 1) ? dur0 : (LAPSE_N - 1)];
    float coef = fmaf(-SA, qm, P);                  // P - SA*qm
    float r    = (1.0f - qm) * (1.0f - lf);

    float  bef = pc;
    float2 de0 = s_de[0];
    float  extra0 = (dm0 == 0) ? (EXPENSE_ACQ + P) * pc : 0.0f; // acq expense + commission
    acc = fmaf(bef, fmaf(coef, de0.x, -de0.y), -extra0 * de0.x);
    bef *= r;

    // ---- steady months, chunked by policy-year ----
    int t  = 1;
    int dm = dm0 + 1;
    while (t < t_end) {
      int dur = dm / 12;
      int dc  = dur < 5 ? dur : 5;
      qm   = s_mort[(age0 + dur - 18 - dc) * MORT_COLS + dc];
      lf   = s_lapse[dur < (LAPSE_N - 1) ? dur : (LAPSE_N - 1)];
      coef = fmaf(-SA, qm, P);
      r    = (1.0f - qm) * (1.0f - lf);

      int months = 12 - (dm % 12);                  // until next policy-year
      int chunk  = t + months < t_end ? months : (t_end - t);
      for (int k = 0; k < chunk; ++k) {
        float2 d2 = s_de[t];
        acc = fmaf(bef, fmaf(coef, d2.x, -d2.y), acc);
        bef *= r;
        ++t;
      }
      dm += chunk;
    }
  }

  // deterministic block tree reduction
  s_red[threadIdx.x] = acc;
  __syncthreads();
  for (int s = 128; s > 0; s >>= 1) {
    if (threadIdx.x < s) s_red[threadIdx.x] += s_red[threadIdx.x + s];
    __syncthreads();
  }
  if (threadIdx.x == 0) partials[blockIdx.x] = s_red[0];
}

// --------------------------------------------------------------------------
// Kernel 3: single-wave final reduction of per-block partials using
// V_WMMA_F32_16X16X4_F32 with an all-ones B matrix:
//   D[m][n] = sum_k A[m][k] + C[m][n]  (row sums broadcast across n)
// Total = column-0 sum of D = lane0{vgpr0..7} + lane16{vgpr0..7}.
// Deterministic: fixed lane->index mapping, fixed summation order.
// --------------------------------------------------------------------------
__global__ void reduce_wmma_kernel(const float* __restrict__ partials,
                                   int nblocks,
                                   float* __restrict__ out) {
  __shared__ float s_col[32];
  int lane = threadIdx.x;

  float ax = 0.0f, ay = 0.0f;
  for (int k = lane; k < nblocks; k += 64)      ax += partials[k];
  for (int k = lane + 32; k < nblocks; k += 64) ay += partials[k];

  v2f a; a.x = ax; a.y = ay;
  v2f b; b.x = 1.0f; b.y = 1.0f;
  v8f c = {};
  // 8 args: (neg_a, A, neg_b, B, c_mod, C, reuse_a, reuse_b)
  c = __builtin_amdgcn_wmma_f32_16x16x4_f32(
      false, a, false, b, (short)0, c, false, false);

  float s = c[0] + c[1] + c[2] + c[3] + c[4] + c[5] + c[6] + c[7];
  s_col[lane] = s;
  __syncthreads();
  if (lane == 0) out[0] = s_col[0] + s_col[16];
}

// --------------------------------------------------------------------------
extern "C" void kernel_launch(void* const* d_in, const int* in_sizes, int n_in,
                              void* d_out, int out_size, void* d_ws, size_t ws_size,
                              hipStream_t stream) {
  const float* premium_pp    = (const float*)d_in[0];
  const float* sum_assured   = (const float*)d_in[1];
  const float* policy_count  = (const float*)d_in[2];
  const float* mort_table    = (const float*)d_in[3];
  const float* disc_rate_ann = (const float*)d_in[4];
  const int*   duration_mth  = (const int*)d_in[5];
  const int*   age_at_entry  = (const int*)d_in[6];
  const int*   policy_term   = (const int*)d_in[7];

  int n = in_sizes[0];
  float* ws = (float*)d_ws;
  float* partials = ws + PART_OFF;
  float* out = (float*)d_out;

  int nblocks = (n + 255) / 256;

  tables_kernel<<<1, 256, 0, stream>>>(mort_table, disc_rate_ann, ws);
  project_kernel<<<nblocks, 256, 0, stream>>>(premium_pp, sum_assured, policy_count,
                                              duration_mth, age_at_entry, policy_term,
                                              ws, partials, n);
  reduce_wmma_kernel<<<1, 32, 0, stream>>>(partials, nblocks, out);
}